// VectorQuantizer_12970801234460
// MI455X (gfx1250) — compile-verified
//
#include <hip/hip_runtime.h>
#include <math.h>

#define DIMS      64
#define KCODES    1024
#define NTOK      (32 * 64 * 64)          // 131072 tokens
#define TILES     (NTOK / 16)             // 8192 16-token tiles
#define CB_PITCH  66                      // padded LDS pitch (words): 64 data + 2 pad
#define K2_BLOCKS ((NTOK * 16) / 256)     // 8192 blocks, one float4 per thread

typedef __attribute__((ext_vector_type(2))) float v2f;
typedef __attribute__((ext_vector_type(8))) float v8f;
typedef __attribute__((ext_vector_type(4))) unsigned int u32x4;
typedef __attribute__((ext_vector_type(8))) int          i32x8;
typedef __attribute__((ext_vector_type(4))) int          i32x4;

// ---------------------------------------------------------------------------
// K0: zero the histogram (workspace is poisoned once and never re-poisoned)
// ---------------------------------------------------------------------------
__global__ void vq_zero_kernel(unsigned int* __restrict__ counts) {
    counts[threadIdx.x] = 0u;
}

// ---------------------------------------------------------------------------
// K1: nearest-codebook-entry search via V_WMMA_F32_16X16X4_F32.
//   - Codebook (256KB, padded pitch 66) + ||e||^2 staged in LDS.
//     Staging uses the Tensor Data Mover (6-arg clang-23 builtin): one 2-D D#
//     (1024x64 f32) with pad_enable so the DMA engine itself lays out the
//     [1024][66] padded image; tracked with TENSORcnt.
//   - Each wave owns a 16-token tile and walks the 64 code-tiles TWO at a
//     time with independent accumulators (acc0/acc1 share the A fragments),
//     so the matrix pipe sees two interleaved WMMA chains instead of one
//     serial chain (we only fit 2 waves/SIMD because of the LDS footprint).
//   - distance = ||e||^2 - 2*x.e (||x||^2 is row-constant: drops out of
//     argmin). Final argmin via shfl_xor over each 16-lane half (C layout:
//     reg r, half h -> row M = r + 8h, N = lane&15), first-index tiebreak.
// ---------------------------------------------------------------------------
__global__ __launch_bounds__(256) void vq_argmin_kernel(
    const float* __restrict__ inp, const float* __restrict__ weight,
    float* __restrict__ out_idx)
{
    extern __shared__ float smem[];
    float* cb  = smem;                      // [1024][66]
    float* esq = smem + KCODES * CB_PITCH;  // [1024]

    const int tid = threadIdx.x;

#if defined(__gfx1250__) && __has_builtin(__builtin_amdgcn_tensor_load_to_lds)
    // ---- TDM codebook load: one DMA op per workgroup (wave 0 issues it) ----
    if (tid < 32) {
        const unsigned long long ga = (unsigned long long)(uintptr_t)weight;
        // D# group 0: count=1(valid), lds_addr=0 (dynamic LDS base),
        // global_addr, type=2 ("image")
        u32x4 g0;
        g0[0] = 1u;                                  // count=1, no gather
        g0[1] = 0u;                                  // lds_addr (bytes)
        g0[2] = (unsigned int)(ga & 0xffffffffull);  // global_addr[31:0]
        g0[3] = (unsigned int)((ga >> 32) & 0x01ffffffull) | (2u << 30); // [56:32] | type=2
        // D# group 1: data_size=4B(2), pad_enable=1, pad_interval=5 (64 DW),
        // pad_amount=1 (2 DW) -> reproduces CB_PITCH=66.
        i32x8 g1;
        g1[0] = (int)((2u << 16) | (1u << 20) | (5u << 22) | (1u << 25));
        g1[1] = (int)(64u << 16);      // tensor_dim0[15:0]=64 at bits 63:48
        g1[2] = (int)(1024u << 16);    // tensor_dim1[15:0]=1024 at bits 95:80
        g1[3] = (int)(64u << 16);      // tile_dim0=64 at bits 127:112
        g1[4] = (int)1024;             // tile_dim1=1024 at bits 143:128 (tile_dim2=0)
        g1[5] = (int)64;               // tensor_dim0_stride[31:0]=64
        g1[6] = 0;                     // stride hi / dim1_stride (2-D: unused)
        g1[7] = 0;
        const i32x4 gz4 = {0, 0, 0, 0};               // groups 2/3: zero (2-D)
        const i32x8 gz8 = {0, 0, 0, 0, 0, 0, 0, 0};   // trailing group: zero
        __builtin_amdgcn_tensor_load_to_lds(g0, g1, gz4, gz4, gz8, 0);
        __builtin_amdgcn_s_wait_tensorcnt(0);
    }
#else
    // ---- Fallback: cooperative float4 global reads -> padded LDS ----
    const float4* w4 = reinterpret_cast<const float4*>(weight);
    for (int i = tid; i < KCODES * (DIMS / 4); i += 256) {
        const int k = i >> 4, c4 = i & 15;
        const float4 v = w4[i];
        float* p = &cb[k * CB_PITCH + 4 * c4];
        p[0] = v.x; p[1] = v.y; p[2] = v.z; p[3] = v.w;
    }
#endif
    __syncthreads();

    // ||e_k||^2 (conflict-free: bank = (2k+d) % 64, distinct across 32 lanes)
    for (int k = tid; k < KCODES; k += 256) {
        const float* p = &cb[k * CB_PITCH];
        float s = 0.f;
        #pragma unroll
        for (int d = 0; d < DIMS; ++d) s += p[d] * p[d];
        esq[k] = s;
    }
    __syncthreads();

    const int lane = tid & 31;
    const int wv   = tid >> 5;
    const int half = lane >> 4;   // K-pair select within a WMMA step
    const int nloc = lane & 15;   // M (for A) / N (for B) within tile
    const int slot   = blockIdx.x * 8 + wv;
    const int nslots = gridDim.x * 8;

    for (int tile = slot; tile < TILES; tile += nslots) {
        // A fragments: ISA layout VGPR0 = K{0,2}, VGPR1 = K{1,3} -> each lane
        // reads a contiguous float2 at column 4c + 2*half of its token row.
        v2f a[16];
        const v2f* rowp =
            reinterpret_cast<const v2f*>(inp + (size_t)(tile * 16 + nloc) * DIMS);
        #pragma unroll
        for (int c = 0; c < 16; ++c) a[c] = rowp[2 * c + half];

        float minv[8];
        int   mini[8];
        #pragma unroll
        for (int r = 0; r < 8; ++r) { minv[r] = 3.4e38f; mini[r] = 0; }

        // Two code-tiles per iteration: independent WMMA chains acc0/acc1.
        for (int nt = 0; nt < KCODES / 16; nt += 2) {
            const int   n0  = nt * 16 + nloc;
            const float eq0 = esq[n0];
            const float eq1 = esq[n0 + 16];
            const float* cbrow0 = &cb[n0 * CB_PITCH + 2 * half];
            const float* cbrow1 = cbrow0 + 16 * CB_PITCH;

            v8f acc0 = {}, acc1 = {};
            #pragma unroll
            for (int c = 0; c < 16; ++c) {
                const v2f b0 = *reinterpret_cast<const v2f*>(cbrow0 + 4 * c);
                const v2f b1 = *reinterpret_cast<const v2f*>(cbrow1 + 4 * c);
                acc0 = __builtin_amdgcn_wmma_f32_16x16x4_f32(
                    false, a[c], false, b0, (short)0, acc0, false, false);
                acc1 = __builtin_amdgcn_wmma_f32_16x16x4_f32(
                    false, a[c], false, b1, (short)0, acc1, false, false);
            }

            #pragma unroll
            for (int r = 0; r < 8; ++r) {
                const float d0 = fmaf(-2.0f, acc0[r], eq0);
                if (d0 < minv[r]) { minv[r] = d0; mini[r] = n0; }
            }
            #pragma unroll
            for (int r = 0; r < 8; ++r) {
                const float d1 = fmaf(-2.0f, acc1[r], eq1);
                if (d1 < minv[r]) { minv[r] = d1; mini[r] = n0 + 16; }
            }
        }

        // Reduce (min, argmin) across the 16 lanes of each half; masks 1..8
        // never cross the half boundary. Ties -> smaller index (jnp.argmin).
        #pragma unroll
        for (int r = 0; r < 8; ++r) {
            float v = minv[r];
            int   i = mini[r];
            #pragma unroll
            for (int m = 1; m <= 8; m <<= 1) {
                const float ov = __shfl_xor(v, m, 32);
                const int   oi = __shfl_xor(i, m, 32);
                if (ov < v || (ov == v && oi < i)) { v = ov; i = oi; }
            }
            if (nloc == 0)
                out_idx[tile * 16 + r + 8 * half] = (float)i;  // exact for i < 2^24
        }
    }
}

// ---------------------------------------------------------------------------
// K2: gather quantized rows (== quantized_st), per-block SSE partial sums
// (deterministic tree reduce), integer histogram atomics. One float4/thread,
// fully coalesced on inputs and outputs.
// ---------------------------------------------------------------------------
__global__ __launch_bounds__(256) void vq_gather_kernel(
    const float* __restrict__ inp, const float* __restrict__ weight,
    const float* __restrict__ idxf, float* __restrict__ out_q,
    unsigned int* __restrict__ counts, float* __restrict__ partials)
{
    __shared__ float red[256];
    const int gid = blockIdx.x * 256 + threadIdx.x;  // float4 index
    const int tok = gid >> 4, c4 = gid & 15;
    const int idx = (int)idxf[tok];

    const float4* w4 = reinterpret_cast<const float4*>(weight);
    const float4* x4 = reinterpret_cast<const float4*>(inp);
    float4*       o4 = reinterpret_cast<float4*>(out_q);

    const float4 w = w4[idx * 16 + c4];
    const float4 x = x4[gid];
    o4[gid] = w;

    const float dx = w.x - x.x, dy = w.y - x.y, dz = w.z - x.z, dw = w.w - x.w;
    red[threadIdx.x] = dx * dx + dy * dy + dz * dz + dw * dw;

    if (c4 == 0) atomicAdd(&counts[idx], 1u);

    __syncthreads();
    #pragma unroll
    for (int off = 128; off > 0; off >>= 1) {
        if (threadIdx.x < off) red[threadIdx.x] += red[threadIdx.x + off];
        __syncthreads();
    }
    if (threadIdx.x == 0) partials[blockIdx.x] = red[0];
}

// ---------------------------------------------------------------------------
// K3: fixed-order final reductions -> vq_loss and perplexity scalars.
// ---------------------------------------------------------------------------
__global__ __launch_bounds__(1024) void vq_finalize_kernel(
    const unsigned int* __restrict__ counts, const float* __restrict__ partials,
    int npart, float* __restrict__ out_loss, float* __restrict__ out_ppl)
{
    __shared__ float rH[1024];
    __shared__ float rS[1024];
    const int t = threadIdx.x;

    const float p = (float)counts[t] * (1.0f / (float)NTOK);
    rH[t] = p * logf(p + 1e-10f);

    float s = 0.f;
    for (int j = t; j < npart; j += 1024) s += partials[j];
    rS[t] = s;

    __syncthreads();
    #pragma unroll
    for (int off = 512; off > 0; off >>= 1) {
        if (t < off) { rH[t] += rH[t + off]; rS[t] += rS[t + off]; }
        __syncthreads();
    }
    if (t == 0) {
        *out_ppl  = expf(-rH[0]);
        // e_latent == q_latent numerically -> (1 + 0.25) * MSE
        *out_loss = 1.25f * rS[0] / (float)(NTOK * DIMS);
    }
}

// ---------------------------------------------------------------------------
extern "C" void kernel_launch(void* const* d_in, const int* in_sizes, int n_in,
                              void* d_out, int out_size, void* d_ws, size_t ws_size,
                              hipStream_t stream) {
    (void)in_sizes; (void)n_in; (void)out_size; (void)ws_size;

    const float* inp    = (const float*)d_in[0];   // [32,64,64,64] f32
    const float* weight = (const float*)d_in[1];   // [1024,64]     f32

    float* out      = (float*)d_out;
    float* out_q    = out;                          // 8388608 elems
    float* out_loss = out + 8388608;                // 1
    float* out_idx  = out + 8388609;                // 131072 (indices as f32)
    float* out_ppl  = out + 8388609 + NTOK;         // 1

    unsigned int* counts   = (unsigned int*)d_ws;                      // 4 KB
    float*        partials = (float*)((char*)d_ws + 4096);             // 32 KB

    vq_zero_kernel<<<1, 1024, 0, stream>>>(counts);

    const size_t lds_bytes = (size_t)(KCODES * CB_PITCH + KCODES) * sizeof(float);
    vq_argmin_kernel<<<256, 256, lds_bytes, stream>>>(inp, weight, out_idx);

    vq_gather_kernel<<<K2_BLOCKS, 256, 0, stream>>>(inp, weight, out_idx, out_q,
                                                    counts, partials);

    vq_finalize_kernel<<<1, 1024, 0, stream>>>(counts, partials, K2_BLOCKS,
                                               out_loss, out_ppl);
}